// DICELossMultiClass_23699629539539
// MI455X (gfx1250) — compile-verified
//
#include <hip/hip_runtime.h>
#include <hip/hip_bf16.h>

// Problem constants (reference: B=16, C=8, H=512, W=512)
#define BB 16
#define CC 8
#define HH 512
#define WW 512
#define HW (HH * WW)                 // 262144 pixels per image
#define GROUPS_PER_BATCH (HW / 4)    // 65536 float4/int4 groups per image
#define BLOCKS_PER_BATCH 64
#define GROUPS_PER_BLOCK (GROUPS_PER_BATCH / BLOCKS_PER_BATCH) // 1024
#define THREADS 256
#define ITERS (GROUPS_PER_BLOCK / THREADS)                      // 4

typedef float v2f __attribute__((ext_vector_type(2)));
typedef float v8f __attribute__((ext_vector_type(8)));

// ---------------------------------------------------------------------------
// Stage 1: streaming reduction. Each block reduces 1024 4-pixel groups of one
// batch image into 24 partial sums (num[8], den1[8], den2[8]) written to ws.
// Fully coalesced float4/int4 loads; register accumulators; wave32 shuffles.
// ---------------------------------------------------------------------------
__global__ __launch_bounds__(THREADS) void dice_stage1(
    const float* __restrict__ output,   // [B,C,H,W] f32
    const int*   __restrict__ mask,     // [B,H,W]   i32 labels 0..7
    float*       __restrict__ partials) // [1024][24]
{
  const int bi   = blockIdx.x;          // 0..1023
  const int b    = bi >> 6;             // batch
  const int blk  = bi & 63;             // sub-block within batch
  const int tid  = threadIdx.x;

  const float4* __restrict__ o4 = reinterpret_cast<const float4*>(output);
  const int4*   __restrict__ m4 = reinterpret_cast<const int4*>(mask);

  const size_t obase = (size_t)b * CC * GROUPS_PER_BATCH; // float4 units
  const size_t mbase = (size_t)b * GROUPS_PER_BATCH;      // int4 units
  const int    g0    = blk * GROUPS_PER_BLOCK + tid;

  float num[CC];
  float den1[CC];
  int   cnt[CC];
#pragma unroll
  for (int c = 0; c < CC; ++c) { num[c] = 0.f; den1[c] = 0.f; cnt[c] = 0; }

#pragma unroll
  for (int i = 0; i < ITERS; ++i) {
    const int g = g0 + i * THREADS;
    const int4 ml = m4[mbase + g];
#pragma unroll
    for (int c = 0; c < CC; ++c) {
      const float4 x = o4[obase + (size_t)c * GROUPS_PER_BATCH + g];
      den1[c] += x.x * x.x + x.y * x.y + x.z * x.z + x.w * x.w;
      float na = (ml.x == c ? x.x : 0.f) + (ml.y == c ? x.y : 0.f)
               + (ml.z == c ? x.z : 0.f) + (ml.w == c ? x.w : 0.f);
      int   ca = (ml.x == c) + (ml.y == c) + (ml.z == c) + (ml.w == c);
      num[c] += na;
      cnt[c] += ca;
    }
  }

  // Wave32 butterfly reduction.
#pragma unroll
  for (int off = 16; off > 0; off >>= 1) {
#pragma unroll
    for (int c = 0; c < CC; ++c) {
      num[c]  += __shfl_xor(num[c],  off);
      den1[c] += __shfl_xor(den1[c], off);
      cnt[c]  += __shfl_xor(cnt[c],  off);
    }
  }

  // Cross-wave reduction through LDS (8 waves per block).
  __shared__ float sp[8][24];
  const int wave = tid >> 5;
  const int lane = tid & 31;
  if (lane == 0) {
#pragma unroll
    for (int c = 0; c < CC; ++c) {
      sp[wave][c]      = num[c];
      sp[wave][8 + c]  = den1[c];
      sp[wave][16 + c] = (float)cnt[c];
    }
  }
  __syncthreads();
  if (tid < 24) {
    float s = 0.f;
#pragma unroll
    for (int w = 0; w < 8; ++w) s += sp[w][tid];
    partials[(size_t)bi * 24 + tid] = s;
  }
}

// ---------------------------------------------------------------------------
// Stage 2: deterministic final reduction + dice + loss.
// Threads 0..383 each own one (b, j) accumulator and sum 64 block-partials in
// fixed order. Wave 0 then computes the 128 dice values and sums them with
// V_WMMA_F32_16X16X4_F32 (A = dice values, B = ones -> exact f32 row sums).
// ---------------------------------------------------------------------------
__global__ __launch_bounds__(384) void dice_stage2(
    const float* __restrict__ partials, // [1024][24]
    float*       __restrict__ out)      // [1] loss
{
  __shared__ float s_acc[BB * 24];      // [b][j], j = kind*8 + c
  const int t = threadIdx.x;            // 0..383
  const int b = t / 24;
  const int j = t % 24;
  {
    const float* p = partials + (size_t)b * BLOCKS_PER_BATCH * 24 + j;
    float s = 0.f;
#pragma unroll
    for (int k = 0; k < BLOCKS_PER_BATCH; ++k) s += p[(size_t)k * 24];
    s_acc[t] = s;
  }
  __syncthreads();

  if (t < 32) {                         // wave 0 only; EXEC all ones inside
    const int lane = t;
    const float eps = 1e-07f;
    float d[4];
#pragma unroll
    for (int q = 0; q < 4; ++q) {
      const int idx = lane + 32 * q;    // (b,c) flat index 0..127
      const int bb = idx >> 3;
      const int cc = idx & 7;
      const float nm = s_acc[bb * 24 + cc];
      const float d1 = s_acc[bb * 24 + 8 + cc];
      const float d2 = s_acc[bb * 24 + 16 + cc];
      d[q] = 2.0f * ((nm + eps) / (d1 + d2 + eps));
    }

    // Sum the 128 dice values with two chained f32 WMMAs (B = ones).
    v2f a0; a0[0] = d[0]; a0[1] = d[1];   // covers flat indices   0..63
    v2f a1; a1[0] = d[2]; a1[1] = d[3];   // covers flat indices 64..127
    v2f ones; ones[0] = 1.0f; ones[1] = 1.0f;
    v8f acc = {};
    acc = __builtin_amdgcn_wmma_f32_16x16x4_f32(false, a1, false, ones,
                                                (short)0, acc, false, false);
    acc = __builtin_amdgcn_wmma_f32_16x16x4_f32(false, a0, false, ones,
                                                (short)0, acc, false, false);
    // D[m][n] = rowsum_m; lanes 0-15 hold rows 0-7 in VGPRs 0-7,
    // lanes 16-31 hold rows 8-15.
    float partial = acc[0] + acc[1] + acc[2] + acc[3] +
                    acc[4] + acc[5] + acc[6] + acc[7];
    partial += __shfl_xor(partial, 16);   // total over all 16 rows

    if (lane == 0) {
      // loss = 1 - sum(dice) / B / B
      out[0] = 1.0f - partial * (1.0f / (float)(BB * BB));
    }
  }
}

// ---------------------------------------------------------------------------
// kernel_launch: inputs per reference order: output [B,C,H,W] f32, mask i32.
// d_ws needs 1024*24*4 = 96 KiB for block partials (written then read; no
// init required, no atomics -> deterministic across graph replays).
// ---------------------------------------------------------------------------
extern "C" void kernel_launch(void* const* d_in, const int* in_sizes, int n_in,
                              void* d_out, int out_size, void* d_ws, size_t ws_size,
                              hipStream_t stream) {
  const float* output = (const float*)d_in[0];
  const int*   mask   = (const int*)d_in[1];
  float*       outp   = (float*)d_out;
  float*       parts  = (float*)d_ws;

  dice_stage1<<<BB * BLOCKS_PER_BATCH, THREADS, 0, stream>>>(output, mask, parts);
  dice_stage2<<<1, 384, 0, stream>>>(parts, outp);
}